// GNNAutoencoder_4398046511871
// MI455X (gfx1250) — compile-verified
//
#include <hip/hip_runtime.h>

// GNN autoencoder forward for MI455X (gfx1250), f32 end-to-end.
// Per-layer GEMMs: V_WMMA_F32_16X16X4_F32, 16x(NT*16) tile per wave, W slab
// staged into LDS via GLOBAL_LOAD_ASYNC_TO_LDS_B32 (ASYNCcnt) once per block.
// Edge aggregation via global f32 atomics (L2-resident working set).

#define NV    50000
#define NE    800000
#define NLAT  128
#define EPS_BN 1e-5f
#define SLOPE  0.01f

typedef __attribute__((ext_vector_type(2))) float v2f;
typedef __attribute__((ext_vector_type(8))) float v8f;

// ---- CDNA5 async to-LDS helpers ----
static __device__ __forceinline__ void async_load_f32_to_lds(unsigned int ldsAddr,
                                                             const float* g) {
  asm volatile("global_load_async_to_lds_b32 %0, %1, off"
               :: "v"(ldsAddr), "v"((unsigned long long)(size_t)g)
               : "memory");
}
static __device__ __forceinline__ void wait_async0() {
#if __has_builtin(__builtin_amdgcn_s_wait_asynccnt)
  __builtin_amdgcn_s_wait_asynccnt(0);
#else
  asm volatile("s_wait_asynccnt 0" ::: "memory");
#endif
}

// ---------------- small utility kernels ----------------

__global__ void fill_kernel(float* __restrict__ p, float v, int n) {
  int i = blockIdx.x * blockDim.x + threadIdx.x;
  if (i < n) p[i] = v;
}

__global__ void deg_accum_kernel(const int* __restrict__ dst, float* __restrict__ deg, int e) {
  int i = blockIdx.x * blockDim.x + threadIdx.x;
  if (i < e) atomicAdd(&deg[dst[i]], 1.0f);
}

__global__ void rsqrt_inplace_kernel(float* __restrict__ p, int n) {
  int i = blockIdx.x * blockDim.x + threadIdx.x;
  if (i < n) p[i] = rsqrtf(p[i]);   // deg >= 1 always (self loop)
}

__global__ void edge_norm_kernel(const int* __restrict__ src, const int* __restrict__ dst,
                                 const float* __restrict__ dinv, float* __restrict__ norm, int e) {
  int i = blockIdx.x * blockDim.x + threadIdx.x;
  if (i < e) norm[i] = dinv[src[i]] * dinv[dst[i]];
}

// ---------------- WMMA f32 GEMM: H[N,cout] = X[N,cin] @ W[cin,cout] ----------------
// Block = 256 threads = 8 waves; wave w owns M-tile (blockIdx.x*8+w), computing a
// 16 x (NT*16) output tile with NT accumulators. The block's W slab
// (cin x NT*16 floats) is staged into LDS with async-to-LDS loads.
// A layout (ISA 7.12.2): lanes 0-15 -> rows M=lane, K={0,1}; lanes 16-31 -> K={2,3}.
// B symmetric (cols across lanes). C/D: VGPR r -> row r (+8 for lanes 16-31).

template <int NT>
__global__ void gemm_wmma_kernel(const float* __restrict__ X, const float* __restrict__ W,
                                 float* __restrict__ H, int n, int cin, int cout) {
  constexpr int SW = NT * 16;                 // slab width (columns)
  __shared__ float slab[256 * SW];            // up to 256 K-rows (max cin) x SW

  const int lane = threadIdx.x & 31;
  const int wv   = (int)threadIdx.x >> 5;
  const int colBase = blockIdx.y * SW;

  // --- stage W[k, colBase..colBase+SW) into LDS asynchronously ---
  {
    const int total = cin * SW;
    const unsigned int ldsBase = (unsigned int)(size_t)slab;  // low 32b of shared aperture = LDS offset
    for (int idx = threadIdx.x; idx < total; idx += 256) {
      int k = idx / SW;
      int c = idx - k * SW;
      async_load_f32_to_lds(ldsBase + (unsigned int)idx * 4u, W + (size_t)k * cout + colBase + c);
    }
    wait_async0();
  }
  __syncthreads();   // all waves reach this (no early return before it)

  const int mTiles = n >> 4;
  const int mTile  = blockIdx.x * 8 + wv;
  if (mTile < mTiles) {                       // wave-uniform: EXEC all-ones at WMMA
    const int l15   = lane & 15;
    const int kHalf = (lane >> 4) << 1;       // 0 or 2
    const float* xrow = X + (size_t)(mTile * 16 + l15) * cin;

    v8f acc[NT];
    #pragma unroll
    for (int t = 0; t < NT; ++t) acc[t] = (v8f){};

    const int kFull = cin & ~3;
    for (int k = 0; k < kFull; k += 4) {
      int k0 = k + kHalf;
      v2f a;
      a.x = xrow[k0];
      a.y = xrow[k0 + 1];
      #pragma unroll
      for (int t = 0; t < NT; ++t) {
        v2f b;
        b.x = slab[k0 * SW + t * 16 + l15];
        b.y = slab[(k0 + 1) * SW + t * 16 + l15];
        acc[t] = __builtin_amdgcn_wmma_f32_16x16x4_f32(false, a, false, b, (short)0,
                                                       acc[t], false, false);
      }
    }
    if (kFull < cin) {                        // zero-padded K tail (layer 0: cin == 1)
      int k0 = kFull + kHalf;
      v2f a;
      a.x = (k0     < cin) ? xrow[k0]     : 0.f;
      a.y = (k0 + 1 < cin) ? xrow[k0 + 1] : 0.f;
      #pragma unroll
      for (int t = 0; t < NT; ++t) {
        v2f b;
        b.x = (k0     < cin) ? slab[k0 * SW + t * 16 + l15]       : 0.f;
        b.y = (k0 + 1 < cin) ? slab[(k0 + 1) * SW + t * 16 + l15] : 0.f;
        acc[t] = __builtin_amdgcn_wmma_f32_16x16x4_f32(false, a, false, b, (short)0,
                                                       acc[t], false, false);
      }
    }

    const int rowBase = mTile * 16 + ((lane >> 4) << 3);
    #pragma unroll
    for (int t = 0; t < NT; ++t) {
      float* hp = H + (size_t)rowBase * cout + colBase + t * 16 + l15;
      #pragma unroll
      for (int r = 0; r < 8; ++r) hp[(size_t)r * cout] = acc[t][r];
    }
  }
}

// ---------------- aggregation ----------------
// agg[i,c] = H[i,c]*dinv[i]^2 + bias[c]   (self-loop + conv bias)
__global__ void agg_init_kernel(const float* __restrict__ H, const float* __restrict__ dinv,
                                const float* __restrict__ bias, float* __restrict__ agg,
                                int total, int C) {
  int i = blockIdx.x * blockDim.x + threadIdx.x;
  if (i < total) {
    int node = i / C;
    float di = dinv[node];
    agg[i] = H[i] * di * di + bias[i & (C - 1)];
  }
}

// agg[dst,c] += H[src,c]*norm[e]; 4 channels per thread, edge shared by consecutive threads
__global__ void agg_scatter_kernel(const int* __restrict__ src, const int* __restrict__ dst,
                                   const float* __restrict__ norm, const float* __restrict__ H,
                                   float* __restrict__ agg, int e, int c4, int C) {
  int t = blockIdx.x * blockDim.x + threadIdx.x;
  if (t >= e * c4) return;
  int ed = t / c4;
  int cc = (t - ed * c4) << 2;
  int s = src[ed], d = dst[ed];
  float w = norm[ed];
  float4 h = *reinterpret_cast<const float4*>(H + (size_t)s * C + cc);
  float* a = agg + (size_t)d * C + cc;
  atomicAdd(a + 0, h.x * w);
  atomicAdd(a + 1, h.y * w);
  atomicAdd(a + 2, h.z * w);
  atomicAdd(a + 3, h.w * w);
}

// ---------------- BatchNorm (training mode, biased variance) ----------------
// blockDim must be 256; C is a power of two <= 256, so each thread owns one fixed channel.
__global__ void bn_stats_kernel(const float* __restrict__ y, float* __restrict__ sums,
                                float* __restrict__ sumsq, int total, int C) {
  __shared__ float ssum[256], ssq[256];
  int tid = threadIdx.x;
  int stride = gridDim.x * 256;
  float ls = 0.f, lq = 0.f;
  for (int i = blockIdx.x * 256 + tid; i < total; i += stride) {
    float v = y[i];
    ls += v; lq += v * v;
  }
  ssum[tid] = ls; ssq[tid] = lq;
  __syncthreads();
  for (int off = 128; off >= C; off >>= 1) {
    if (tid < off) { ssum[tid] += ssum[tid + off]; ssq[tid] += ssq[tid + off]; }
    __syncthreads();
  }
  if (tid < C) {
    atomicAdd(&sums[tid],  ssum[tid]);
    atomicAdd(&sumsq[tid], ssq[tid]);
  }
}

__global__ void bn_finalize_kernel(const float* __restrict__ sums, const float* __restrict__ sumsq,
                                   const float* __restrict__ g, const float* __restrict__ b,
                                   float* __restrict__ scale, float* __restrict__ shift,
                                   float invn, int C) {
  int c = blockIdx.x * blockDim.x + threadIdx.x;
  if (c < C) {
    float m  = sums[c] * invn;
    float v  = fmaxf(sumsq[c] * invn - m * m, 0.f);
    float sc = g[c] * rsqrtf(v + EPS_BN);
    scale[c] = sc;
    shift[c] = b[c] - m * sc;
  }
}

__global__ void bn_apply_kernel(const float* __restrict__ y, const float* __restrict__ scale,
                                const float* __restrict__ shift, float* __restrict__ x,
                                int total, int C, int doLeaky) {
  int i = blockIdx.x * blockDim.x + threadIdx.x;
  if (i < total) {
    int c = i & (C - 1);
    float v = y[i] * scale[c] + shift[c];
    if (doLeaky) v = (v >= 0.f) ? v : SLOPE * v;
    x[i] = v;
  }
}

// ---------------- head: pool -> fc -> output ----------------
__global__ void fc_kernel(const float* __restrict__ sums, const float* __restrict__ fw,
                          const float* __restrict__ fb, float* __restrict__ latent, float invn) {
  __shared__ float sp[256];
  int j = threadIdx.x;                      // blockDim == 128
  sp[j]       = sums[j]       * invn;
  sp[j + 128] = sums[j + 128] * invn;
  __syncthreads();
  float acc = fb[j];
  for (int k = 0; k < 256; ++k) acc += sp[k] * fw[k * NLAT + j];
  latent[j] = acc;
}

__global__ void out_kernel(const float* __restrict__ latent, const float* __restrict__ ow,
                           float* __restrict__ out, int nv) {
  __shared__ float sl[NLAT];
  if (threadIdx.x < NLAT) sl[threadIdx.x] = latent[threadIdx.x];
  __syncthreads();
  int j = blockIdx.x * blockDim.x + threadIdx.x;
  if (j < nv) {
    float acc = 0.f;
    for (int k = 0; k < NLAT; ++k) acc += sl[k] * ow[(size_t)k * nv + j];
    out[j] = acc;
  }
}

// ---------------- host launcher ----------------

extern "C" void kernel_launch(void* const* d_in, const int* in_sizes, int n_in,
                              void* d_out, int out_size, void* d_ws, size_t ws_size,
                              hipStream_t stream) {
  (void)in_sizes; (void)out_size; (void)ws_size;
  const int N = NV, E = NE;
  const int chans[11] = {1, 16, 16, 32, 32, 64, 64, 128, 128, 256, 256};

  const float* x_in = (const float*)d_in[0];
  const int*   ei   = (const int*)d_in[1];
  const int*   src  = ei;         // edge_index[0]: message sources
  const int*   dst  = ei + E;     // edge_index[1]: aggregation targets

  const float *conv_w[10], *conv_b[10], *bn_g[11], *bn_b[11];
  const float *fc_w, *fc_b, *out_w;
  if (n_in >= 47) {               // lists flattened into separate inputs
    for (int i = 0; i < 10; ++i) { conv_w[i] = (const float*)d_in[2 + i];
                                   conv_b[i] = (const float*)d_in[12 + i]; }
    for (int i = 0; i < 11; ++i) { bn_g[i] = (const float*)d_in[22 + i];
                                   bn_b[i] = (const float*)d_in[33 + i]; }
    fc_w = (const float*)d_in[44]; fc_b = (const float*)d_in[45]; out_w = (const float*)d_in[46];
  } else {                        // lists concatenated per dict entry
    const float* cw = (const float*)d_in[2];
    const float* cb = (const float*)d_in[3];
    const float* bg = (const float*)d_in[4];
    const float* bb = (const float*)d_in[5];
    size_t wo = 0, bo = 0, go = 0;
    for (int i = 0; i < 10; ++i) {
      conv_w[i] = cw + wo; wo += (size_t)chans[i] * chans[i + 1];
      conv_b[i] = cb + bo; bo += chans[i + 1];
    }
    for (int i = 0; i < 11; ++i) { bn_g[i] = bg + go; bn_b[i] = bb + go; go += chans[i]; }
    fc_w = (const float*)d_in[6]; fc_b = (const float*)d_in[7]; out_w = (const float*)d_in[8];
  }

  float* out    = (float*)d_out;  // [N]
  float* latent = out + N;        // [128]

  float* ws   = (float*)d_ws;
  float* bufX = ws;                         // N*256 floats
  float* bufH = ws   + (size_t)N * 256;     // N*256 floats
  float* dinv = bufH + (size_t)N * 256;     // N floats (deg -> dinv in place)
  float* norm = dinv + N;                   // E floats
  float* sums = norm + E;                   // 256
  float* sumsq = sums + 256;                // 256 (contiguous with sums)
  float* scale = sumsq + 256;               // 256
  float* shift = scale + 256;               // 256

  auto cdiv = [](int a, int b) { return (a + b - 1) / b; };

  // GCN symmetric normalization (self loops included): deg, dinv, per-edge norm
  fill_kernel<<<cdiv(N, 256), 256, 0, stream>>>(dinv, 1.0f, N);
  deg_accum_kernel<<<cdiv(E, 256), 256, 0, stream>>>(dst, dinv, E);
  rsqrt_inplace_kernel<<<cdiv(N, 256), 256, 0, stream>>>(dinv, N);
  edge_norm_kernel<<<cdiv(E, 256), 256, 0, stream>>>(src, dst, dinv, norm, E);

  // initial BN (C=1, no LeakyReLU): x0 = BN(input)
  fill_kernel<<<2, 256, 0, stream>>>(sums, 0.f, 512);
  bn_stats_kernel<<<512, 256, 0, stream>>>(x_in, sums, sumsq, N, 1);
  bn_finalize_kernel<<<1, 256, 0, stream>>>(sums, sumsq, bn_g[0], bn_b[0], scale, shift, 1.f / N, 1);
  bn_apply_kernel<<<cdiv(N, 256), 256, 0, stream>>>(x_in, scale, shift, bufX, N, 1, 0);

  const int mBlocks = cdiv(N / 16, 8);      // 8 M-tiles (waves) per block
  for (int l = 0; l < 10; ++l) {
    int cin = chans[l], cout = chans[l + 1];
    // H = X @ W (WMMA f32, LDS-staged W slab)
    if (cout >= 64) {
      gemm_wmma_kernel<4><<<dim3(mBlocks, cout / 64), 256, 0, stream>>>(bufX, conv_w[l], bufH, N, cin, cout);
    } else if (cout == 32) {
      gemm_wmma_kernel<2><<<dim3(mBlocks, 1), 256, 0, stream>>>(bufX, conv_w[l], bufH, N, cin, cout);
    } else {
      gemm_wmma_kernel<1><<<dim3(mBlocks, 1), 256, 0, stream>>>(bufX, conv_w[l], bufH, N, cin, cout);
    }
    // agg (in old X buffer): self-loop + bias, then edge scatter
    agg_init_kernel<<<cdiv(N * cout, 256), 256, 0, stream>>>(bufH, dinv, conv_b[l], bufX, N * cout, cout);
    agg_scatter_kernel<<<cdiv(E * (cout / 4), 256), 256, 0, stream>>>(src, dst, norm, bufH, bufX, E, cout / 4, cout);
    // BN + LeakyReLU -> new X (in old H buffer)
    fill_kernel<<<2, 256, 0, stream>>>(sums, 0.f, 512);
    bn_stats_kernel<<<1024, 256, 0, stream>>>(bufX, sums, sumsq, N * cout, cout);
    bn_finalize_kernel<<<1, 256, 0, stream>>>(sums, sumsq, bn_g[l + 1], bn_b[l + 1], scale, shift, 1.f / N, cout);
    bn_apply_kernel<<<cdiv(N * cout, 256), 256, 0, stream>>>(bufX, scale, shift, bufH, N * cout, cout, 1);
    float* t = bufX; bufX = bufH; bufH = t;
  }

  // global mean pool (sum) -> fc -> output
  fill_kernel<<<2, 256, 0, stream>>>(sums, 0.f, 512);
  bn_stats_kernel<<<1024, 256, 0, stream>>>(bufX, sums, sumsq, N * 256, 256);
  fc_kernel<<<1, 128, 0, stream>>>(sums, fc_w, fc_b, latent, 1.f / N);
  out_kernel<<<cdiv(N, 256), 256, 0, stream>>>(latent, out_w, out, N);
}